// Neuron_56143812493595
// MI455X (gfx1250) — compile-verified
//
#include <hip/hip_runtime.h>

typedef __attribute__((ext_vector_type(2))) float v2f;
typedef __attribute__((ext_vector_type(8))) float v8f;

constexpr int   kB        = 65536;  // batch rows
constexpr int   kNN       = 2048;   // NUM_NEURONS (row stride of neuron_inputs)
constexpr int   kFanIn    = 128;    // dot length; columns are k*16
constexpr float kNegSlope = 0.01f;

// One wave32 per 16-row tile. Dot product done with exact-fp32 WMMA:
//   D(16x16) += A(16x4) * B(4x16), 32 chained K-chunks cover K=128.
// B holds the weight chunk replicated across all 16 N columns, so every
// column of C equals input_vec @ weights for its row.
__global__ __launch_bounds__(256) void neuron_tile16_wmma_f32(
    const float* __restrict__ nin,    // (B, 2048) neuron_inputs
    const float* __restrict__ prev,   // (B,)      prev_state
    const float* __restrict__ w,      // (128,)    weights
    const float* __restrict__ bias,   // (1,)
    const float* __restrict__ alpha,  // (1,)
    float* __restrict__ out)          // (2*B,) [new_state | output]
{
    const int lane = threadIdx.x & 31;
    const int wave = threadIdx.x >> 5;
    const int tile = blockIdx.x * (blockDim.x >> 5) + wave;
    const int row0 = tile << 4;                // 16 rows per tile

    const int half = lane >> 4;                // 0: lanes 0-15, 1: lanes 16-31
    const int m16  = lane & 15;                // A-matrix row M within tile

    const float* __restrict__ rowp = nin + (size_t)(row0 + m16) * kNN;

    v8f c = {};                                // fp32 accumulator (C/D 16x16)

    // A 16x4 f32 layout: VGPR0 = K=(half*2), VGPR1 = K+1, M = lane&15.
    // B 4x16 f32, replicated over N: lanes 0-15 carry w[4t],w[4t+1];
    // lanes 16-31 carry w[4t+2],w[4t+3].
#pragma unroll
    for (int t = 0; t < kFanIn / 4; ++t) {
        const int k0 = (t << 2) + (half << 1);
        v2f a, b;
        a.x = rowp[(size_t)k0 * 16];           // column = k*16 (stride 64B)
        a.y = rowp[(size_t)(k0 + 1) * 16];
        b.x = w[k0];
        b.y = w[k0 + 1];
        // 8 args: (neg_a, A, neg_b, B, c_mod, C, reuse_a, reuse_b)
        c = __builtin_amdgcn_wmma_f32_16x16x4_f32(
                false, a, false, b, (short)0, c, false, false);
    }

    // C layout: VGPR i -> M=i (lanes 0-15), M=8+i (lanes 16-31); all N equal.
    // Lanes 0-7 own rows 0-7, lanes 16-23 own rows 8-15.
    const int j = lane & 7;
    float dot = c[0];
#pragma unroll
    for (int i = 1; i < 8; ++i) dot = (j == i) ? c[i] : dot;

    if (m16 < 8) {                             // divergence only after all WMMAs
        const int row = row0 + (half << 3) + j;
        const float decay = tanhf(alpha[0] * prev[row]);
        const float total = dot + bias[0] + decay;
        const float ns    = (total > 0.0f) ? total : kNegSlope * total;
        out[row]          = ns;                // new_state
        out[kB + row]     = tanhf(total);      // output
    }
}

extern "C" void kernel_launch(void* const* d_in, const int* in_sizes, int n_in,
                              void* d_out, int out_size, void* d_ws, size_t ws_size,
                              hipStream_t stream) {
    // Input order from setup_inputs():
    //   0: model_inputs (unused by reference), 1: neuron_inputs,
    //   2: prev_state, 3: weights, 4: bias, 5: alpha
    const float* nin   = (const float*)d_in[1];
    const float* prev  = (const float*)d_in[2];
    const float* w     = (const float*)d_in[3];
    const float* bias  = (const float*)d_in[4];
    const float* alpha = (const float*)d_in[5];
    float* out = (float*)d_out;

    const int tiles          = kB / 16;        // 4096 waves
    const int wavesPerBlock  = 8;              // 256 threads
    dim3 grid(tiles / wavesPerBlock);          // 512 blocks
    dim3 block(wavesPerBlock * 32);

    neuron_tile16_wmma_f32<<<grid, block, 0, stream>>>(nin, prev, w, bias, alpha, out);
}